// KalmanFilter_54374285967481
// MI455X (gfx1250) — compile-verified
//
#include <hip/hip_runtime.h>

// Kalman filter, G=1024 groups x T=200 sequential steps, S=16, M=4 (all f32).
// One wave32 per group; 16x16 covariance matmuls on V_WMMA_F32_16X16X4_F32.

#define G_ 1024
#define T_ 200
#define S_ 16
#define M_ 4

typedef __attribute__((ext_vector_type(2))) float v2f;
typedef __attribute__((ext_vector_type(8))) float v8f;

__device__ __forceinline__ v8f wmma_f32(v2f a, v2f b, v8f c) {
  // D = A(16x4) * B(4x16) + C(16x16), f32. 8-arg form:
  // (neg_a, A, neg_b, B, c_mod, C, reuse_a, reuse_b)
  return __builtin_amdgcn_wmma_f32_16x16x4_f32(false, a, false, b,
                                               (short)0, c, false, false);
}

__global__ __launch_bounds__(32) void kalman_step_chain(
    const float* __restrict__ y,         // [G,T,M]
    const float* __restrict__ F,         // [G,T,S,S]
    const float* __restrict__ H,         // [G,T,M,S]
    const float* __restrict__ Q,         // [G,T,S,S]
    const float* __restrict__ R,         // [G,T,M,M]
    const float* __restrict__ init_mean, // [G,S]
    const float* __restrict__ init_cov,  // [G,S,S]
    float* __restrict__ out_means,       // [G,T,S]
    float* __restrict__ out_covs)        // [G,T,S,S]
{
  __shared__ float sF[S_ * S_];
  __shared__ float sCov[S_ * S_];
  __shared__ float sCovU[S_ * S_];
  __shared__ float sT1[S_ * S_];
  __shared__ float sH[M_ * S_];
  __shared__ float sHP[M_ * S_];
  __shared__ float sK[S_ * M_];   // holds -K (16x4), row-major
  __shared__ float sSsig[M_ * M_];
  __shared__ float sMean[S_];
  __shared__ float sMeanU[S_];
  __shared__ float sResid[M_];

  const int g    = blockIdx.x;
  const int lane = threadIdx.x;   // 0..31 (wave32)
  const int m    = lane & 15;     // N / row-within-half index
  const int h    = lane >> 4;     // half-wave select

  // ---- init state ----
  if (h == 0) sMean[m] = init_mean[(size_t)g * S_ + m];
#pragma unroll
  for (int r = 0; r < 8; ++r)
    sCov[lane * 8 + r] = init_cov[(size_t)g * 256 + lane * 8 + r];
  __syncthreads();

  for (int t = 0; t < T_; ++t) {
    const size_t gt = (size_t)g * T_ + t;
    const float* Ft = F + gt * 256;
    const float* Ht = H + gt * 64;
    const float* Qt = Q + gt * 256;
    const float* Rt = R + gt * 16;
    const float* yt = y + gt * 4;

    // ---- emit prior (1-step-ahead prediction) for time t ----
    if (h == 0)
      __builtin_nontemporal_store(sMean[m], &out_means[gt * S_ + m]);
#pragma unroll
    for (int r = 0; r < 8; ++r)
      __builtin_nontemporal_store(sCov[lane * 8 + r],
                                  &out_covs[gt * 256 + lane * 8 + r]);

    // ---- stage H_t and F_t into LDS ----
    sH[lane]      = Ht[lane];
    sH[lane + 32] = Ht[lane + 32];
    {
      const float4* Fv = (const float4*)Ft;
      ((float4*)sF)[lane]      = Fv[lane];
      ((float4*)sF)[lane + 32] = Fv[lane + 32];
    }
    if (t + 1 < T_) {  // pull next step's big operands toward the WGP
      __builtin_prefetch(Ft + 256, 0, 1);
      __builtin_prefetch(Qt + 256, 0, 1);
    }
    __syncthreads();

    // ---- (1) resid = y - H*mean  (lanes 0..3, one element each) ----
    if (lane < M_) {
      float acc = yt[lane];
#pragma unroll
      for (int k = 0; k < S_; ++k) acc -= sH[lane * S_ + k] * sMean[k];
      sResid[lane] = acc;
    }

    // ---- (2) HP = H @ cov  (lane -> rows {2h,2h+1}, col m) ----
    {
      const int i0 = 2 * h, i1 = i0 + 1;
      float a0 = 0.f, a1 = 0.f;
#pragma unroll
      for (int k = 0; k < S_; ++k) {
        const float c = sCov[k * S_ + m];
        a0 += sH[i0 * S_ + k] * c;
        a1 += sH[i1 * S_ + k] * c;
      }
      sHP[i0 * S_ + m] = a0;
      sHP[i1 * S_ + m] = a1;
    }
    __syncthreads();

    // ---- (3) Ssig = HP*H^T + R  (lanes 0..15, one element each) ----
    if (h == 0) {
      const int i = m >> 2, j = m & 3;
      float acc = Rt[i * M_ + j];
#pragma unroll
      for (int k = 0; k < S_; ++k) acc += sHP[i * S_ + k] * sH[j * S_ + k];
      sSsig[m] = acc;
    }
    __syncthreads();

    // ---- (4) K = (Ssig^-1 HP)^T : redundant 4x4 Cholesky, per-col solve ----
    {
      const float s00 = sSsig[0],  s11 = sSsig[5],  s22 = sSsig[10], s33 = sSsig[15];
      const float s10 = sSsig[4],  s20 = sSsig[8],  s30 = sSsig[12];
      const float s21 = sSsig[9],  s31 = sSsig[13], s32 = sSsig[14];
      const float d0  = sqrtf(fmaxf(s00, 1e-30f));
      const float L10 = s10 / d0, L20 = s20 / d0, L30 = s30 / d0;
      const float d1  = sqrtf(fmaxf(s11 - L10 * L10, 1e-30f));
      const float L21 = (s21 - L20 * L10) / d1;
      const float L31 = (s31 - L30 * L10) / d1;
      const float d2  = sqrtf(fmaxf(s22 - L20 * L20 - L21 * L21, 1e-30f));
      const float L32 = (s32 - L30 * L20 - L31 * L21) / d2;
      const float d3  = sqrtf(fmaxf(s33 - L30 * L30 - L31 * L31 - L32 * L32, 1e-30f));
      // column m of HP
      const float b0 = sHP[0 * S_ + m], b1 = sHP[1 * S_ + m];
      const float b2 = sHP[2 * S_ + m], b3 = sHP[3 * S_ + m];
      // forward: L z = b
      const float z0 = b0 / d0;
      const float z1 = (b1 - L10 * z0) / d1;
      const float z2 = (b2 - L20 * z0 - L21 * z1) / d2;
      const float z3 = (b3 - L30 * z0 - L31 * z1 - L32 * z2) / d3;
      // backward: L^T x = z   (x = K[m][0..3])
      const float x3 = z3 / d3;
      const float x2 = (z2 - L32 * x3) / d2;
      const float x1 = (z1 - L21 * x2 - L31 * x3) / d1;
      const float x0 = (z0 - L10 * x1 - L20 * x2 - L30 * x3) / d0;
      if (h == 0) {  // store -K (f32 WMMA has no A-negate modifier)
        sK[m * M_ + 0] = -x0; sK[m * M_ + 1] = -x1;
        sK[m * M_ + 2] = -x2; sK[m * M_ + 3] = -x3;
      }
    }
    __syncthreads();

    // ---- (5) mean_u = mean + K*resid  (lanes 0..15) ----
    if (h == 0) {
      float acc = sMean[m];
#pragma unroll
      for (int i = 0; i < M_; ++i) acc -= sK[m * M_ + i] * sResid[i]; // sK = -K
      sMeanU[m] = acc;
    }
    __syncthreads();

    // ---- (6) ImKH = (-K)@H + I : single 16x16x4 WMMA ----
    {
      v8f eye;
#pragma unroll
      for (int r = 0; r < 8; ++r) eye[r] = ((r + 8 * h) == m) ? 1.0f : 0.0f;
      v2f a, b;
      a.x = sK[m * M_ + 2 * h + 0];
      a.y = sK[m * M_ + 2 * h + 1];
      b.x = sH[(2 * h + 0) * S_ + m];
      b.y = sH[(2 * h + 1) * S_ + m];
      const v8f imkh = wmma_f32(a, b, eye);
#pragma unroll
      for (int r = 0; r < 8; ++r) sT1[(r + 8 * h) * S_ + m] = imkh[r];
    }
    __syncthreads();

    // ---- (7) cov_u = ImKH @ cov : 4 chained WMMAs ----
    {
      v8f cu = {0.f, 0.f, 0.f, 0.f, 0.f, 0.f, 0.f, 0.f};
#pragma unroll
      for (int k0 = 0; k0 < S_; k0 += 4) {
        v2f a, b;
        a.x = sT1[m * S_ + k0 + 2 * h + 0];
        a.y = sT1[m * S_ + k0 + 2 * h + 1];
        b.x = sCov[(k0 + 2 * h + 0) * S_ + m];
        b.y = sCov[(k0 + 2 * h + 1) * S_ + m];
        cu = wmma_f32(a, b, cu);
      }
#pragma unroll
      for (int r = 0; r < 8; ++r) sCovU[(r + 8 * h) * S_ + m] = cu[r];
    }
    __syncthreads();

    // ---- (8) T1 = F @ cov_u : 4 chained WMMAs ----
    {
      v8f t1 = {0.f, 0.f, 0.f, 0.f, 0.f, 0.f, 0.f, 0.f};
#pragma unroll
      for (int k0 = 0; k0 < S_; k0 += 4) {
        v2f a, b;
        a.x = sF[m * S_ + k0 + 2 * h + 0];
        a.y = sF[m * S_ + k0 + 2 * h + 1];
        b.x = sCovU[(k0 + 2 * h + 0) * S_ + m];
        b.y = sCovU[(k0 + 2 * h + 1) * S_ + m];
        t1 = wmma_f32(a, b, t1);
      }
#pragma unroll
      for (int r = 0; r < 8; ++r) sT1[(r + 8 * h) * S_ + m] = t1[r];
    }
    __syncthreads();

    // ---- (9) cov_p = T1 @ F^T + Q : C seeded with Q fragment ----
    v8f cp;
#pragma unroll
    for (int r = 0; r < 8; ++r) cp[r] = Qt[(r + 8 * h) * S_ + m];
#pragma unroll
    for (int k0 = 0; k0 < S_; k0 += 4) {
      v2f a, b;
      a.x = sT1[m * S_ + k0 + 2 * h + 0];
      a.y = sT1[m * S_ + k0 + 2 * h + 1];
      b.x = sF[m * S_ + k0 + 2 * h + 0];   // B-frag of F^T: F^T[k][m] = F[m][k]
      b.y = sF[m * S_ + k0 + 2 * h + 1];
      cp = wmma_f32(a, b, cp);
    }

    // ---- (10) mean_p = F @ mean_u (lanes 0..15); commit new state ----
    if (h == 0) {
      float acc = 0.f;
#pragma unroll
      for (int k = 0; k < S_; ++k) acc += sF[m * S_ + k] * sMeanU[k];
      sMean[m] = acc;  // old sMean readers all finished before earlier barriers
    }
#pragma unroll
    for (int r = 0; r < 8; ++r) sCov[(r + 8 * h) * S_ + m] = cp[r];
    __syncthreads();
  }
}

extern "C" void kernel_launch(void* const* d_in, const int* in_sizes, int n_in,
                              void* d_out, int out_size, void* d_ws, size_t ws_size,
                              hipStream_t stream) {
  const float* y         = (const float*)d_in[0];
  const float* F         = (const float*)d_in[1];
  const float* H         = (const float*)d_in[2];
  const float* Q         = (const float*)d_in[3];
  const float* R         = (const float*)d_in[4];
  const float* init_mean = (const float*)d_in[5];
  const float* init_cov  = (const float*)d_in[6];

  float* out       = (float*)d_out;
  const size_t GT  = (size_t)G_ * T_;
  float* out_means = out;                          // [G,T,S]
  float* out_covs  = out + GT * S_;                // [G,T,S,S]
  float* out_R     = out_covs + GT * S_ * S_;      // [G,T,M,M] (verbatim copy)
  float* out_H     = out_R + GT * M_ * M_;         // [G,T,M,S] (verbatim copy)

  // R and H are returned unchanged: pure d2d copies (graph-capture safe).
  hipMemcpyAsync(out_R, R, GT * M_ * M_ * sizeof(float),
                 hipMemcpyDeviceToDevice, stream);
  hipMemcpyAsync(out_H, H, GT * M_ * S_ * sizeof(float),
                 hipMemcpyDeviceToDevice, stream);

  kalman_step_chain<<<G_, 32, 0, stream>>>(y, F, H, Q, R, init_mean, init_cov,
                                           out_means, out_covs);
}